// Model_76879914598797
// MI455X (gfx1250) — compile-verified
//
#include <hip/hip_runtime.h>

// Problem constants (match reference)
#define NA   10000      // nodes
#define NE   80000      // edges
#define NB   100        // batch segments
#define HD   128        // hidden
#define KR   20         // rbf freqs
#define NL   3          // layers
#define CUT  5.0f
#define PI_F 3.14159265358979f

typedef float v2f __attribute__((ext_vector_type(2)));
typedef float v8f __attribute__((ext_vector_type(8)));

// ---------------------------------------------------------------------------
// Fused GEMM:  C[M,Nd] = prelu(A[M,Kd] @ W[Kd,Nd] + bias, slope)
//
// fp32 WMMA 16x16x4.  Block = 4 waves = 64x64 C tile; wave w owns rows
// 16w..16w+15 with 4 accumulators (cols n0..n0+63) so each A fragment is
// reused across 4 WMMAs.  K consumed in chunks of 8 (all K dims here are
// multiples of 8).
//
// Data movement: CDNA5 async global->LDS copies (GLOBAL_LOAD_ASYNC_TO_LDS_
// B128, ASYNCcnt) with DOUBLE-BUFFERED LDS tiles.  Each wave issues exactly
// 2 async-copy instructions per chunk; chunk c+1 is issued into the alternate
// buffer *before* waiting, then `s_wait_asynccnt 2` retires only the current
// chunk's copies (async loads complete in order per wave), overlapping the
// prefetch with the 8 WMMAs.  The trailing barrier of iteration c-1
// guarantees buffer (c+1)&1 is no longer being read when the prefetch lands.
//
// LDS: As[2][64][12] (8 used cols; 48B rows keep 16B-aligned b128 chunks and
// conflict-free column reads: 12*k mod 64 distinct for k=0..15),
// Bs[2][8][64] (row-contiguous, conflict-free).  ~10KB / 320KB WGP.
//
// Requires Kd % 8 == 0, Nd % 64 == 0.  M arbitrary (clamp loads, guard
// stores; bounds branches are wave-uniform or post-WMMA so EXEC is all-ones
// at every WMMA).
// ---------------------------------------------------------------------------
__global__ void k_gemm_wmma(const float* __restrict__ A,
                            const float* __restrict__ W,
                            const float* __restrict__ bias,
                            const float* __restrict__ slope,
                            float* __restrict__ C,
                            int M, int Kd, int Nd)
{
    __shared__ float As[2][64][12];
    __shared__ float Bs[2][8][64];

    const int t    = threadIdx.x;        // 0..127
    const int lane = t & 31;
    const int wave = t >> 5;
    const int m0   = blockIdx.x << 6;    // 64-row block
    const int n0   = blockIdx.y << 6;    // 64-col block

    const int mr = lane & 15;            // row in 16x16 tile (A) / col (B,C)
    const int kh = (lane >> 4) << 1;     // 0 (lanes 0-15) or 2 (lanes 16-31)

    // per-thread async-copy assignments (one b128 = 4 floats each)
    const int ar = t >> 1;               // A: row 0..63
    const int ap = (t & 1) << 2;         // A: col 0 or 4
    const int br = t >> 4;               // B: row 0..7
    const int bc = (t & 15) << 2;        // B: col 0,4,...,60

    int arow_g = m0 + ar; if (arow_g >= M) arow_g = M - 1;   // clamp (no OOB)

    const uint32_t ldsA0 = (uint32_t)(size_t)(&As[0][ar][ap]);
    const uint32_t ldsA1 = (uint32_t)(size_t)(&As[1][ar][ap]);
    const uint32_t ldsB0 = (uint32_t)(size_t)(&Bs[0][br][bc]);
    const uint32_t ldsB1 = (uint32_t)(size_t)(&Bs[1][br][bc]);

    const float* gA = A + (size_t)arow_g * Kd + ap;          // chunk c: +8c
    const float* gB = W + (size_t)br * Nd + (n0 + bc);       // chunk c: +8c*Nd

    v8f acc[4] = {};
    const int nch = Kd >> 3;

    // prologue: chunk 0 -> buffer 0
    asm volatile("global_load_async_to_lds_b128 %0, %1, off"
                 :: "v"(ldsA0), "v"(gA) : "memory");
    asm volatile("global_load_async_to_lds_b128 %0, %1, off"
                 :: "v"(ldsB0), "v"(gB) : "memory");

    for (int c = 0; c < nch; ++c) {
        const int buf = c & 1;
        if (c + 1 < nch) {
            // prefetch next chunk into alternate buffer, then deferred wait:
            // <=2 outstanding ==> current chunk's 2 copies have landed.
            const float* ga = gA + ((c + 1) << 3);
            const float* gb = gB + (size_t)((c + 1) << 3) * Nd;
            asm volatile("global_load_async_to_lds_b128 %0, %1, off"
                         :: "v"(buf ? ldsA0 : ldsA1), "v"(ga) : "memory");
            asm volatile("global_load_async_to_lds_b128 %0, %1, off"
                         :: "v"(buf ? ldsB0 : ldsB1), "v"(gb) : "memory");
            asm volatile("s_wait_asynccnt 0x2" ::: "memory");
        } else {
            asm volatile("s_wait_asynccnt 0x0" ::: "memory");
        }
        __syncthreads();

        const float* arow_s = &As[buf][(wave << 4) + mr][0];
#pragma unroll
        for (int kk = 0; kk < 8; kk += 4) {
            v2f a;
            a[0] = arow_s[kk + kh];
            a[1] = arow_s[kk + kh + 1];
#pragma unroll
            for (int j = 0; j < 4; ++j) {
                v2f b;
                b[0] = Bs[buf][kk + kh][(j << 4) + mr];
                b[1] = Bs[buf][kk + kh + 1][(j << 4) + mr];
                acc[j] = __builtin_amdgcn_wmma_f32_16x16x4_f32(
                             false, a, false, b, (short)0, acc[j], false, false);
            }
        }
        __syncthreads();   // required: next iteration's prefetch overwrites `buf`
    }

    // epilogue: bias + prelu + guarded store
    const float sl = slope[0];
    const int rbase = m0 + (wave << 4) + ((lane >> 4) << 3);
#pragma unroll
    for (int j = 0; j < 4; ++j) {
        const int col = n0 + (j << 4) + mr;
        const float bv = bias[col];
#pragma unroll
        for (int r = 0; r < 8; ++r) {
            const int row = rbase + r;
            if (row < M) {
                float x = acc[j][r] + bv;
                x = (x >= 0.f) ? x : sl * x;
                C[(size_t)row * Nd + col] = x;
            }
        }
    }
}

// Naive linear+prelu for shapes WMMA can't tile (K=13 embed, tiny FLOPs).
__global__ void k_linear_naive(const float* __restrict__ A,
                               const float* __restrict__ W,
                               const float* __restrict__ bias,
                               const float* __restrict__ slope,
                               float* __restrict__ C,
                               int M, int Kd, int Nd)
{
    int idx = blockIdx.x * blockDim.x + threadIdx.x;
    if (idx >= M * Nd) return;
    int row = idx / Nd, col = idx % Nd;
    float acc = bias[col];
    for (int k = 0; k < Kd; ++k) acc += A[row * Kd + k] * W[k * Nd + col];
    float sl = slope[0];
    C[idx] = (acc >= 0.f) ? acc : sl * acc;
}

__global__ void k_zero(float* __restrict__ p, int n)
{
    int i = blockIdx.x * blockDim.x + threadIdx.x;
    int s = gridDim.x * blockDim.x;
    for (; i < n; i += s) p[i] = 0.f;
}

__global__ void k_add(float* __restrict__ a, const float* __restrict__ b, int n)
{
    int i = blockIdx.x * blockDim.x + threadIdx.x;
    int s = gridDim.x * blockDim.x;
    for (; i < n; i += s) a[i] += b[i];
}

// rbf_dist[E, 2*KR]: cols [0,20) from bonds[:,0], [20,40) from bonds[:,1]
__global__ void k_rbf(const float* __restrict__ bonds, float* __restrict__ rbf)
{
    int idx = blockIdx.x * blockDim.x + threadIdx.x;
    if (idx >= NE * 2 * KR) return;
    int e = idx / (2 * KR);
    int t = idx % (2 * KR);
    int j = t / KR;
    int k = (t % KR) + 1;
    float d = bonds[e * 2 + j];
    float r = 0.f;
    if (d < CUT) r = sinf(d * (float)k * PI_F / CUT) / d;
    rbf[idx] = r * 0.5f * (cosf(PI_F * r / CUT) + 1.f);
}

// Fused edge message pass.  One block per edge, one thread per channel h.
// fo = scalar_out[src] * filt;  (g_sv, m_s, g_ev) = split(fo)
// ds[dst]      += m_s
// dv[dst,d,h]  += nv[src,d,h]*g_sv + unit_d*g_ev
// Deltas accumulated separately so the gather of nv[src] sees pre-update values.
__global__ void k_edge_scatter(const int*   __restrict__ pairs,
                               const float* __restrict__ diff,
                               const float* __restrict__ dist,
                               const float* __restrict__ filt,        // [E, 384]
                               const float* __restrict__ scalar_out,  // [N, 384]
                               const float* __restrict__ nv,          // [N, 3, 128]
                               float* __restrict__ ds,                // [N, 128]
                               float* __restrict__ dv)                // [N, 3, 128]
{
    const int e = blockIdx.x;
    const int h = threadIdx.x;
    const int dst = pairs[2 * e];
    const int src = pairs[2 * e + 1];
    const size_t eb = (size_t)e * 3 * HD;
    const size_t sb = (size_t)src * 3 * HD;
    const float g_sv = scalar_out[sb + h]            * filt[eb + h];
    const float m_s  = scalar_out[sb + HD + h]       * filt[eb + HD + h];
    const float g_ev = scalar_out[sb + 2 * HD + h]   * filt[eb + 2 * HD + h];
    atomicAdd(&ds[(size_t)dst * HD + h], m_s);
    const float invd = 1.0f / dist[e];
#pragma unroll
    for (int d = 0; d < 3; ++d) {
        const float unit = diff[e * 3 + d] * invd;
        const float m = nv[sb + (size_t)d * HD + h] * g_sv + unit * g_ev;
        atomicAdd(&dv[(size_t)dst * 3 * HD + (size_t)d * HD + h], m);
    }
}

// cat[i, 0:128] = ||v[i,:,h]||, cat[i, 128:256] = ns[i,h]
__global__ void k_vnorm_concat(const float* __restrict__ v,
                               const float* __restrict__ ns,
                               float* __restrict__ cat)
{
    int idx = blockIdx.x * blockDim.x + threadIdx.x;
    if (idx >= NA * HD) return;
    int i = idx / HD, h = idx % HD;
    size_t vb = (size_t)i * 3 * HD + h;
    float a = v[vb], b = v[vb + HD], c = v[vb + 2 * HD];
    cat[(size_t)i * 2 * HD + h]      = sqrtf(a * a + b * b + c * c);
    cat[(size_t)i * 2 * HD + HD + h] = ns[idx];
}

// ns += sum_d(u*v)*a_sv + a_ss ;  nv += u * a_vv
__global__ void k_update_apply(const float* __restrict__ mo,   // [N, 384]
                               const float* __restrict__ u,    // [N, 3, 128]
                               const float* __restrict__ v,    // [N, 3, 128]
                               float* __restrict__ ns,
                               float* __restrict__ nv)
{
    int idx = blockIdx.x * blockDim.x + threadIdx.x;
    if (idx >= NA * HD) return;
    int i = idx / HD, h = idx % HD;
    size_t mb = (size_t)i * 3 * HD;
    float a_vv = mo[mb + h];
    float a_sv = mo[mb + HD + h];
    float a_ss = mo[mb + 2 * HD + h];
    float s = 0.f;
#pragma unroll
    for (int d = 0; d < 3; ++d)
        s += u[mb + (size_t)d * HD + h] * v[mb + (size_t)d * HD + h];
    ns[idx] += s * a_sv + a_ss;
#pragma unroll
    for (int d = 0; d < 3; ++d)
        nv[mb + (size_t)d * HD + h] += u[mb + (size_t)d * HD + h] * a_vv;
}

__global__ void k_seg_offsets(const int* __restrict__ num_atoms, int* __restrict__ offs)
{
    if (threadIdx.x == 0 && blockIdx.x == 0) {
        int acc = 0;
        for (int b = 0; b < NB; ++b) { offs[b] = acc; acc += num_atoms[b]; }
        offs[NB] = acc;
    }
}

// per-node: dot(ns64, w) + b  -> atomicAdd into segment sum
__global__ void k_readout_final(const float* __restrict__ ns64,  // [N, 64]
                                const float* __restrict__ w,     // [64, 1]
                                const float* __restrict__ b,     // [1]
                                const int*   __restrict__ offs,  // [NB+1]
                                float* __restrict__ out)
{
    int i = blockIdx.x * blockDim.x + threadIdx.x;
    if (i >= NA) return;
    float acc = b[0];
    for (int k = 0; k < 64; ++k) acc += ns64[(size_t)i * 64 + k] * w[k];
    int seg = 0;
    for (int s = 0; s < NB; ++s) if (offs[s] <= i) seg = s;
    atomicAdd(&out[seg], acc);
}

// ---------------------------------------------------------------------------
// Host orchestration.
// Param leaf order: jax pytree flattening (dict keys sorted):
//   0..5   atoms_embed, pairs, diff, distance, bonds, num_atoms
//   6..11  emb:      w0(13x64) b0 a0, w1(64x128) b1 a1
//   12+12l msg[l]:   filt w0(40x128) b0 a0, w1(128x384) b1 a1,
//                    smlp w0(128x128) b0 a0, w1(128x384) b1 a1
//   48..56 readout:  3 x (w b a)                (128x128, 128x128, 128x64)
//   57..58 readout_final: w(64x1), b(1)
//   59+12l upd[l]:   mlp w0(256x128) b0 a0, w1(128x384) b1 a1,
//                    u w(128x128) b a, v w(128x128) b a
// ---------------------------------------------------------------------------
extern "C" void kernel_launch(void* const* d_in, const int* in_sizes, int n_in,
                              void* d_out, int out_size, void* d_ws, size_t ws_size,
                              hipStream_t stream)
{
    const float* atoms     = (const float*)d_in[0];
    const int*   pairs     = (const int*)  d_in[1];
    const float* diff      = (const float*)d_in[2];
    const float* dist      = (const float*)d_in[3];
    const float* bonds     = (const float*)d_in[4];
    const int*   num_atoms = (const int*)  d_in[5];
    auto P = [&](int i) { return (const float*)d_in[i]; };

    // workspace carve-up (floats)
    float* w    = (float*)d_ws;
    float* ns   = w;  w += NA * HD;          // node_scalar           [N,128]
    float* nv   = w;  w += NA * 3 * HD;      // node_vector           [N,3,128]
    float* tna  = w;  w += NA * HD;          // hidden / ds-delta     [N,128]
    float* tnb  = w;  w += NA * 3 * HD;      // scalar_out / mo       [N,384]
    float* ubuf = w;  w += NA * 3 * HD;      // dv-delta / u          [N,3,128]
    float* vbuf = w;  w += NA * 3 * HD;      // v                     [N,3,128]
    float* cat  = w;  w += NA * 2 * HD;      // [vnorm | ns]          [N,256]
    float* rbf  = w;  w += NE * 2 * KR;      // rbf_dist              [E,40]
    float* tea  = w;  w += NE * HD;          // filt hidden           [E,128]
    float* teb  = w;  w += NE * 3 * HD;      // filt                  [E,384]
    int*   offs = (int*)w;                   // segment offsets       [NB+1]

    auto gemm = [&](const float* A, int wi, float* C, int M, int Kd, int Nd) {
        dim3 grid((M + 63) / 64, Nd / 64);
        k_gemm_wmma<<<grid, dim3(128), 0, stream>>>(A, P(wi), P(wi + 1), P(wi + 2),
                                                    C, M, Kd, Nd);
    };

    // ---- embedding: 13 -> 64 (naive) -> 128 (wmma) ----
    k_linear_naive<<<(NA * 64 + 127) / 128, 128, 0, stream>>>(
        atoms, P(6), P(7), P(8), tna, NA, 13, 64);
    gemm(tna, 9, ns, NA, 64, HD);
    k_zero<<<2048, 256, 0, stream>>>(nv, NA * 3 * HD);

    // ---- rbf features ----
    k_rbf<<<(NE * 2 * KR + 127) / 128, 128, 0, stream>>>(bonds, rbf);

    // ---- message passing layers ----
    for (int l = 0; l < NL; ++l) {
        const int mb = 12 + 12 * l;   // msg params
        const int ub = 59 + 12 * l;   // upd params

        // scalar_out = mlp(ns)        [N,384]
        gemm(ns,  mb + 6, tna, NA, HD, HD);
        gemm(tna, mb + 9, tnb, NA, HD, 3 * HD);
        // filt = mlp(rbf)             [E,384]   (dominant GEMM: M=80000)
        gemm(rbf, mb + 0, tea, NE, 2 * KR, HD);
        gemm(tea, mb + 3, teb, NE, HD, 3 * HD);

        // fused gather * filter * scatter-add into deltas
        k_zero<<<2048, 256, 0, stream>>>(tna,  NA * HD);
        k_zero<<<2048, 256, 0, stream>>>(ubuf, NA * 3 * HD);
        k_edge_scatter<<<NE, HD, 0, stream>>>(pairs, diff, dist, teb, tnb, nv, tna, ubuf);
        k_add<<<2048, 256, 0, stream>>>(ns, tna,  NA * HD);
        k_add<<<2048, 256, 0, stream>>>(nv, ubuf, NA * 3 * HD);

        // u, v = prelu(nv @ Wu/Wv + b)   treated as (3N,128)x(128,128)
        gemm(nv, ub + 6, ubuf, 3 * NA, HD, HD);
        gemm(nv, ub + 9, vbuf, 3 * NA, HD, HD);

        // mo = mlp([ |v|, ns ])        [N,384]
        k_vnorm_concat<<<(NA * HD + 127) / 128, 128, 0, stream>>>(vbuf, ns, cat);
        gemm(cat, ub + 0, tna, NA, 2 * HD, HD);
        gemm(tna, ub + 3, tnb, NA, HD, 3 * HD);

        k_update_apply<<<(NA * HD + 127) / 128, 128, 0, stream>>>(tnb, ubuf, vbuf, ns, nv);
    }

    // ---- readout: 128 -> 128 -> 128 -> 64 -> 1 -> segment_sum ----
    gemm(ns,  48, tna, NA, HD, HD);
    gemm(tna, 51, tnb, NA, HD, HD);
    gemm(tnb, 54, tna, NA, HD, 64);

    k_seg_offsets<<<1, 32, 0, stream>>>(num_atoms, offs);
    k_zero<<<1, 128, 0, stream>>>((float*)d_out, out_size);
    k_readout_final<<<(NA + 127) / 128, 128, 0, stream>>>(
        tna, P(57), P(58), offs, (float*)d_out);
}